// LinearWeightModel_16982300688474
// MI455X (gfx1250) — compile-verified
//
#include <hip/hip_runtime.h>
#include <math.h>

typedef __attribute__((ext_vector_type(2))) float v2f;
typedef __attribute__((ext_vector_type(8))) float v8f;

#define NPIX   (121*240)     // 29040 pixels (lat*lon)
#define BDIM   16            // batch (WMMA M)
#define MT     16            // n_models*T (WMMA N)
#define CDIM   16            // 2*F_IN    (WMMA K)
#define MENS   11
#define FIN    8
#define HEADSZ ((size_t)BDIM * MT * NPIX)   // 7,434,240 floats per head

// ---------------------------------------------------------------------------
// Stage 1: ensemble mean over 11 members + depth moments (mean, std ddof=1).
// One thread per (b, pix, f): 44 coalesced loads (f fastest across lanes).
// Output layout is WMMA-A-friendly: xws[pix][c][b], c = f (mu) or f+8 (sd),
// so stage-2 A loads are contiguous across the 16 lanes of a half-wave.
// ---------------------------------------------------------------------------
__global__ __launch_bounds__(256) void moments_kernel(
    const float* __restrict__ feat, float* __restrict__ xws)
{
    unsigned g   = blockIdx.x * 256u + threadIdx.x;       // exact grid, no tail
    unsigned f   = g & 7u;
    unsigned bp  = g >> 3;
    unsigned b   = bp / NPIX;
    unsigned pix = bp - b * NPIX;

    // feat index: (((b*11 + m)*NPIX + pix)*4 + d)*8 + f ; 32 floats per (b,m,pix)
    const float* base = feat + ((size_t)b * MENS) * NPIX * 32 + (size_t)pix * 32 + f;
    float s0 = 0.f, s1 = 0.f, s2 = 0.f, s3 = 0.f;
    #pragma unroll
    for (int m = 0; m < MENS; ++m) {
        const float* p = base + (size_t)m * NPIX * 32;
        s0 += p[0]; s1 += p[8]; s2 += p[16]; s3 += p[24];
    }
    const float inv = 1.0f / (float)MENS;
    s0 *= inv; s1 *= inv; s2 *= inv; s3 *= inv;
    float mu = 0.25f * (s0 + s1 + s2 + s3);
    float d0 = s0 - mu, d1 = s1 - mu, d2 = s2 - mu, d3 = s3 - mu;
    float sd = sqrtf((d0*d0 + d1*d1 + d2*d2 + d3*d3) * (1.0f / 3.0f)); // ddof=1

    float* xo = xws + (size_t)pix * 256;   // [pix][c][b]
    xo[f * 16 + b]       = mu;             // c = f
    xo[(f + 8) * 16 + b] = sd;             // c = f + 8
}

// ---------------------------------------------------------------------------
// Stage 2: per-pixel 16x16x16 GEMM via 4 chained V_WMMA_F32_16X16X4_F32.
//   D[b, mt] = sum_c x[b,c] * W[mt,c] + bias[mt]
// A = x (M=b, K=c), B = W^T (K=c, N=mt), C preloaded with bias (broadcast M).
// 8 waves/block, 4 pixels/wave -> 32 consecutive pixels/block. Results staged
// in padded LDS (stride 33, conflict-free) then stored coalesced.
// ---------------------------------------------------------------------------
__global__ __launch_bounds__(256) void heads_wmma_kernel(
    const float* __restrict__ xws,
    const float* __restrict__ w_t2m, const float* __restrict__ b_t2m,
    const float* __restrict__ w_tp,  const float* __restrict__ b_tp,
    float* __restrict__ out)
{
    __shared__ float tile[256 * 33];               // 33,792 B

    const unsigned tid  = threadIdx.x;
    const unsigned lane = tid & 31u;
    const unsigned wv   = tid >> 5;                // wave 0..7
    const unsigned Lm   = lane & 15u;              // M/N index within half-wave
    const unsigned h    = lane >> 4;               // half-wave select (K pairing)
    const unsigned pix0 = blockIdx.x * 32u;

    const float* Wh[2] = { w_t2m, w_tp };
    const float* Vh[2] = { b_t2m, b_tp };

    for (int head = 0; head < 2; ++head) {
        const float* W  = Wh[head];
        const float* Bv = Vh[head];
        #pragma unroll
        for (int p = 0; p < 4; ++p) {
            unsigned j   = wv * 4u + (unsigned)p;  // column slot in LDS tile
            unsigned pix = pix0 + j;               // wave-uniform
            if (pix < NPIX) {
                // C = bias[mt] broadcast along M: every C/D VGPR row holds N=Lm.
                float bias = Bv[(size_t)Lm * NPIX + pix];
                v8f acc;
                #pragma unroll
                for (int i = 0; i < 8; ++i) acc[i] = bias;

                const float* xp = xws + (size_t)pix * 256;  // [c][b] tile
                const float* wp = W   + (size_t)pix * 16;   // + mt*NPIX*16 + c
                #pragma unroll
                for (int k = 0; k < 4; ++k) {
                    // ISA layout: VGPR0 holds K = {4k, 4k+2}, VGPR1 = {4k+1, 4k+3}
                    unsigned c0 = 4u * (unsigned)k + 2u * h;
                    v2f a, bm;
                    a[0]  = xp[c0 * 16 + Lm];               // A[M=Lm][K=c0]
                    a[1]  = xp[(c0 + 1) * 16 + Lm];         // A[M=Lm][K=c0+1]
                    bm[0] = wp[(size_t)Lm * (NPIX * 16) + c0];      // B[K=c0][N=Lm]
                    bm[1] = wp[(size_t)Lm * (NPIX * 16) + c0 + 1];  // B[K=c0+1][N=Lm]
                    acc = __builtin_amdgcn_wmma_f32_16x16x4_f32(
                        false, a, false, bm, (short)0, acc, false, false);
                }
                // D VGPR i, lane L: D[M = i + 8h][N = Lm]
                #pragma unroll
                for (int i = 0; i < 8; ++i) {
                    unsigned row = ((unsigned)i + 8u * h) * 16u + Lm;  // b*16 + mt
                    tile[row * 33u + j] = acc[i];
                }
            }
        }
        __syncthreads();
        // Coalesced writeback: out[head][row = b*16+mt][pix], 32 px per wave row
        float* o = out + (size_t)head * HEADSZ;
        const unsigned col = lane;
        const bool inb = (pix0 + col) < NPIX;
        for (unsigned r = wv; r < 256u; r += 8u) {
            if (inb) o[(size_t)r * NPIX + pix0 + col] = tile[r * 33u + col];
        }
        __syncthreads();
    }
}

// ---------------------------------------------------------------------------
extern "C" void kernel_launch(void* const* d_in, const int* in_sizes, int n_in,
                              void* d_out, int out_size, void* d_ws, size_t ws_size,
                              hipStream_t stream)
{
    (void)in_sizes; (void)n_in; (void)out_size; (void)ws_size;
    const float* feat  = (const float*)d_in[0];
    const float* w_t2m = (const float*)d_in[1];
    const float* b_t2m = (const float*)d_in[2];
    const float* w_tp  = (const float*)d_in[3];
    const float* b_tp  = (const float*)d_in[4];
    float* out = (float*)d_out;
    float* xws = (float*)d_ws;   // needs NPIX*256*4 = 29,736,960 bytes

    // Stage 1: 16*29040*8 = 3,717,120 threads = exactly 14,520 blocks of 256
    moments_kernel<<<(BDIM * NPIX * FIN) / 256, 256, 0, stream>>>(feat, xws);

    // Stage 2: 32 pixels per block
    heads_wmma_kernel<<<(NPIX + 31) / 32, 256, 0, stream>>>(
        xws, w_t2m, b_t2m, w_tp, b_tp, out);
}